// TemporalGraphNetwork_79894981640739
// MI455X (gfx1250) — compile-verified
//
#include <hip/hip_runtime.h>
#include <math.h>

// ---- problem constants (match reference) ----
#define HID   128
#define EVT   4096
#define NNODE 20000
#define NEDGE 320000
#define MAXN  100000
#define MAXE  300000

typedef __attribute__((ext_vector_type(2))) float v2f;
typedef __attribute__((ext_vector_type(8))) float v8f;

__device__ __forceinline__ float sigm(float x) { return 1.0f / (1.0f + __expf(-x)); }
__device__ __forceinline__ float tanh_fast(float x) {
  float e = __expf(-2.0f * x);
  return (1.0f - e) / (1.0f + e);
}

// ======================================================================
// Generic fp32 WMMA GEMM:  C[M,N] = A[M,K] @ W[N,K]^T (+ bias[N])
// One wave (32 lanes) computes a 16 x (16*NT) tile using
// V_WMMA_F32_16X16X4_F32.  A-layout (ISA 7.12.2): lane<16 holds row=lane,
// K=k0..k0+1; lane>=16 holds row=lane-16, K=k0+2..k0+3.  B mirrors with
// N across lanes.  C: vgpr g -> row m0+g (lanes<16) / m0+g+8 (lanes>=16),
// col n0+(lane&15).
// ======================================================================
template <int NT>
__global__ void k_gemm_wmma(const float* __restrict__ A, const float* __restrict__ W,
                            const float* __restrict__ bias, float* __restrict__ C,
                            int M, int N, int K) {
  const int lane = threadIdx.x;            // 0..31
  const int m0 = blockIdx.x * 16;
  const int n0 = blockIdx.y * 16 * NT;
  const int mrow = lane & 15;
  const int koff = (lane < 16) ? 0 : 2;

  v8f acc[NT] = {};
  const float* arow = A + (size_t)(m0 + mrow) * K + koff;

  for (int k0 = 0; k0 < K; k0 += 4) {
    v2f a;
    a.x = arow[k0];
    a.y = arow[k0 + 1];
#pragma unroll
    for (int t = 0; t < NT; ++t) {
      const float* wrow = W + (size_t)(n0 + t * 16 + mrow) * K + k0 + koff;
      v2f b;
      b.x = wrow[0];
      b.y = wrow[1];
      acc[t] = __builtin_amdgcn_wmma_f32_16x16x4_f32(false, a, false, b, (short)0,
                                                     acc[t], false, false);
    }
  }

  const int mtop = (lane < 16) ? 0 : 8;
#pragma unroll
  for (int t = 0; t < NT; ++t) {
    const int n = n0 + t * 16 + mrow;
    const float bi = bias ? bias[n] : 0.0f;
#pragma unroll
    for (int g = 0; g < 8; ++g)
      C[(size_t)(m0 + mtop + g) * N + n] = acc[t][g] + bi;
  }
}

// ======================================================================
// Representative-event map: map[id] = min event index using that id.
// ======================================================================
__global__ void k_assign(const int* __restrict__ src_ids, const int* __restrict__ dst_ids,
                         unsigned* __restrict__ map_src, unsigned* __restrict__ map_dst) {
  int i = blockIdx.x * blockDim.x + threadIdx.x;
  if (i >= EVT) return;
  atomicMin(&map_src[src_ids[i]], (unsigned)i);
  atomicMin(&map_dst[dst_ids[i]], (unsigned)i);
}

// ======================================================================
// Build messages on the fly and scatter-add into compact per-unique-id
// accumulators (row = representative event index).  Also counts.
// ======================================================================
__global__ void k_scatter_msgs(const int* __restrict__ et, const int* __restrict__ src_ids,
                               const float* __restrict__ src_mask, const int* __restrict__ dst_ids,
                               const float* __restrict__ dst_mask, const int* __restrict__ eev,
                               const float* __restrict__ ev_emb, const float* __restrict__ ev_mask,
                               const float* __restrict__ ts, const float* __restrict__ last_update,
                               const float* __restrict__ memory, const float* __restrict__ time_w,
                               const float* __restrict__ time_b,
                               const unsigned* __restrict__ map_src, const unsigned* __restrict__ map_dst,
                               float* __restrict__ agg_src, float* __restrict__ agg_dst,
                               float* __restrict__ cnt_src, float* __restrict__ cnt_dst) {
  const int i = blockIdx.x;   // event
  const int k = threadIdx.x;  // 0..127
  const int s = src_ids[i], d = dst_ids[i];
  const float sm = src_mask[i], dm = dst_mask[i], em = ev_mask[i];
  const float t = ts[i];
  const int typ = et[i];
  const float rel = t - last_update[eev[i]] * dm;
  const float isnode = (typ == 3 || typ == 4) ? 1.0f : 0.0f;
  const float targ = t * isnode + rel * dm;
  const float tse = __cosf(targ * time_w[k] + time_b[k]) * em;  // ts_emb (includes event_mask)
  const float se = memory[(size_t)s * HID + k] * sm;
  const float de = memory[(size_t)d * HID + k] * dm;
  const float te = (float)typ;
  const float ee = ev_emb[(size_t)i * HID + k];

  const int rs = (int)map_src[s];
  const int rd = (int)map_dst[d];
  float* as = agg_src + (size_t)rs * (5 * HID);
  float* ad = agg_dst + (size_t)rd * (5 * HID);

  // src msg = [type, src_emb, dst_emb, ts_emb, ev_emb] * event_mask
  atomicAdd(as + 0 * HID + k, te * em);
  atomicAdd(as + 1 * HID + k, se * em);
  atomicAdd(as + 2 * HID + k, de * em);
  atomicAdd(as + 3 * HID + k, tse * em);
  atomicAdd(as + 4 * HID + k, ee * em);
  // dst msg = [type, dst_emb, src_emb, ts_emb, ev_emb] * dst_mask
  atomicAdd(ad + 0 * HID + k, te * dm);
  atomicAdd(ad + 1 * HID + k, de * dm);
  atomicAdd(ad + 2 * HID + k, se * dm);
  atomicAdd(ad + 3 * HID + k, tse * dm);
  atomicAdd(ad + 4 * HID + k, ee * dm);
  if (k == 0) {
    atomicAdd(cnt_src + rs, 1.0f);  // reference counts all events (masked too)
    atomicAdd(cnt_dst + rd, 1.0f);
  }
}

// ======================================================================
// wcat (512 x 768): rows 0..255 = [w_ih | w_hh] (r,z combined),
// rows 256..383 = [w_ih_n | 0]  (i_n), rows 384..511 = [0 | w_hh_n] (h_n)
// ======================================================================
__global__ void k_build_wcat(const float* __restrict__ w_ih, const float* __restrict__ w_hh,
                             float* __restrict__ wcat) {
  const int r = blockIdx.x;                         // 0..511
  const int col = blockIdx.y * 256 + threadIdx.x;   // 0..767
  float v = 0.0f;
  if (col < 640) {
    if (r < 384) v = w_ih[r * 640 + col];
  } else {
    const int c = col - 640;
    if (r < 256) v = w_hh[r * HID + c];
    else if (r >= 384) v = w_hh[(r - 128) * HID + c];
  }
  wcat[r * 768 + col] = v;
}

// xh rows 0..EVT-1 = src events, EVT..2EVT-1 = dst events.
// cols 0..639 = aggregated message / count, cols 640..767 = old memory[id]
__global__ void k_build_xh(const int* __restrict__ src_ids, const int* __restrict__ dst_ids,
                           const unsigned* __restrict__ map_src, const unsigned* __restrict__ map_dst,
                           const float* __restrict__ agg_src, const float* __restrict__ agg_dst,
                           const float* __restrict__ cnt_src, const float* __restrict__ cnt_dst,
                           const float* __restrict__ memory, float* __restrict__ xh) {
  const int row = blockIdx.x;                        // 0..2*EVT-1
  const int col = blockIdx.y * 256 + threadIdx.x;    // 0..767
  const bool isSrc = row < EVT;
  const int i = isSrc ? row : row - EVT;
  const int id = isSrc ? src_ids[i] : dst_ids[i];
  const int rep = (int)(isSrc ? map_src[id] : map_dst[id]);
  float v;
  if (col < 640) {
    const float c = isSrc ? cnt_src[rep] : cnt_dst[rep];
    const float* agg = isSrc ? agg_src : agg_dst;
    v = agg[(size_t)rep * 640 + col] / fmaxf(c, 1.0f);
  } else {
    v = memory[(size_t)id * HID + (col - 640)];
  }
  xh[(size_t)row * 768 + col] = v;
}

// g layout per row (512): [r+z combined (256) | i_n (128) | h_n (128)]
__global__ void k_gru_gate(const float* __restrict__ g, const float* __restrict__ xh,
                           const float* __restrict__ b_ih, const float* __restrict__ b_hh,
                           float* __restrict__ newv) {
  const int i = blockIdx.x;   // 0..2*EVT-1
  const int k = threadIdx.x;  // 0..127
  const float* gr = g + (size_t)i * 512;
  const float r = sigm(gr[k] + b_ih[k] + b_hh[k]);
  const float z = sigm(gr[128 + k] + b_ih[128 + k] + b_hh[128 + k]);
  const float n = tanh_fast(gr[256 + k] + b_ih[256 + k] + r * (gr[384 + k] + b_hh[256 + k]));
  const float h = xh[(size_t)i * 768 + 640 + k];
  newv[(size_t)i * HID + k] = (1.0f - z) * n + z * h;
}

// duplicate ids write identical values -> racing stores are benign
__global__ void k_scatter_set(const int* __restrict__ ids, const float* __restrict__ vals,
                              float* __restrict__ mem2) {
  const int i = blockIdx.x, k = threadIdx.x;
  mem2[(size_t)ids[i] * HID + k] = vals[(size_t)i * HID + k];
}

__global__ void k_node_x(const int* __restrict__ node_ids, const float* __restrict__ node_features,
                         const float* __restrict__ mem2, float* __restrict__ xnode) {
  const int n = blockIdx.x, k = threadIdx.x;
  const int id = node_ids[n];
  xnode[(size_t)n * HID + k] = node_features[(size_t)id * HID + k] + mem2[(size_t)id * HID + k];
}

__global__ void k_build_wqkvs(const float* __restrict__ wq, const float* __restrict__ bq,
                              const float* __restrict__ wk, const float* __restrict__ bk,
                              const float* __restrict__ wv, const float* __restrict__ bv,
                              const float* __restrict__ wskip, const float* __restrict__ bskip,
                              float* __restrict__ wqkvs, float* __restrict__ bqkvs) {
  const int r = blockIdx.x, c = threadIdx.x;  // 512 x 128
  const int rr = r & 127;
  const float* w;
  const float* b;
  if (r < 128)      { w = wq;    b = bq; }
  else if (r < 256) { w = wk;    b = bk; }
  else if (r < 384) { w = wv;    b = bv; }
  else              { w = wskip; b = bskip; }
  wqkvs[r * HID + c] = w[rr * HID + c];
  if (c == 0) bqkvs[r] = b[rr];
}

// ======================================================================
// Edge projection e = edge_attr @ we^T.  One wave per 16 edges.
// edge_attr (16 x 256) = [cos((time-last_update)*tw+tb) | edge_features]
// built in LDS (row pad 257 to dodge bank conflicts), then 8 WMMA N-tiles.
// ======================================================================
__global__ void k_edge_proj_wmma(const int* __restrict__ edge_ids, const float* __restrict__ last_update,
                                 const float* __restrict__ edge_features, const float* __restrict__ time_w,
                                 const float* __restrict__ time_b, const float* __restrict__ we,
                                 const int* __restrict__ time_scalar, float* __restrict__ eout) {
  __shared__ float attr[16 * 257];
  const int lane = threadIdx.x;  // 0..31
  const int m0 = blockIdx.x * 16;
  const float tnow = (float)time_scalar[0];

  for (int idx = lane; idx < 16 * 256; idx += 32) {
    const int r = idx >> 8;
    const int c = idx & 255;
    const int eid = edge_ids[m0 + r];
    float v;
    if (c < HID) {
      const float rt = tnow - last_update[eid];
      v = __cosf(rt * time_w[c] + time_b[c]);
    } else {
      v = edge_features[(size_t)eid * HID + (c - HID)];
    }
    attr[r * 257 + c] = v;
  }
  __syncthreads();

  const int mrow = lane & 15;
  const int koff = (lane < 16) ? 0 : 2;
  v8f acc[8] = {};

  for (int k0 = 0; k0 < 256; k0 += 4) {
    v2f a;
    a.x = attr[mrow * 257 + k0 + koff];
    a.y = attr[mrow * 257 + k0 + koff + 1];
#pragma unroll
    for (int t = 0; t < 8; ++t) {
      const float* wrow = we + (size_t)(t * 16 + mrow) * 256 + k0 + koff;
      v2f b;
      b.x = wrow[0];
      b.y = wrow[1];
      acc[t] = __builtin_amdgcn_wmma_f32_16x16x4_f32(false, a, false, b, (short)0,
                                                     acc[t], false, false);
    }
  }

  const int mtop = (lane < 16) ? 0 : 8;
#pragma unroll
  for (int t = 0; t < 8; ++t) {
    const int n = t * 16 + mrow;
#pragma unroll
    for (int g = 0; g < 8; ++g)
      eout[(size_t)(m0 + mtop + g) * HID + n] = acc[t][g];
  }
}

// one wave per edge: alpha = q[dst].(k[src]+e) / sqrt(H); ordered-uint atomicMax
__global__ void k_alpha(const int* __restrict__ edge_index, const float* __restrict__ qkvs,
                        const float* __restrict__ eout, float* __restrict__ alpha,
                        unsigned* __restrict__ amax_ord) {
  const int w = threadIdx.x >> 5;
  const int lane = threadIdx.x & 31;
  const int e = blockIdx.x * 8 + w;
  const int s = edge_index[e];
  const int d = edge_index[NEDGE + e];
  float sum = 0.0f;
  for (int k = lane; k < HID; k += 32)
    sum += qkvs[(size_t)d * 512 + k] * (qkvs[(size_t)s * 512 + HID + k] + eout[(size_t)e * HID + k]);
  for (int off = 16; off; off >>= 1) sum += __shfl_xor(sum, off, 32);
  if (lane == 0) {
    const float a = sum * 0.08838834764831845f;  // 1/sqrt(128)
    alpha[e] = a;
    const unsigned bits = __float_as_uint(a);
    const unsigned ord = (bits & 0x80000000u) ? ~bits : (bits | 0x80000000u);
    atomicMax(&amax_ord[d], ord);
  }
}

__global__ void k_exp_denom(const int* __restrict__ edge_index, const float* __restrict__ alpha,
                            const unsigned* __restrict__ amax_ord, float* __restrict__ exv,
                            float* __restrict__ denom) {
  const int e = blockIdx.x * blockDim.x + threadIdx.x;
  if (e >= NEDGE) return;
  const int d = edge_index[NEDGE + e];
  const unsigned ord = amax_ord[d];
  const float amax = (ord & 0x80000000u) ? __uint_as_float(ord & 0x7FFFFFFFu)
                                         : __uint_as_float(~ord);
  const float ex = __expf(alpha[e] - amax);
  exv[e] = ex;
  atomicAdd(&denom[d], ex);
}

__global__ void k_skip_init(const float* __restrict__ qkvs, float* __restrict__ out) {
  const int n = blockIdx.x, k = threadIdx.x;
  out[(size_t)n * HID + k] = qkvs[(size_t)n * 512 + 384 + k];  // x@wskip.T + bskip
}

__global__ void k_out(const int* __restrict__ edge_index, const float* __restrict__ qkvs,
                      const float* __restrict__ eout, const float* __restrict__ exv,
                      const float* __restrict__ denom, float* __restrict__ out) {
  const int e = blockIdx.x, k = threadIdx.x;
  const int s = edge_index[e];
  const int d = edge_index[NEDGE + e];
  const float p = exv[e] / fmaxf(denom[d], 1e-16f);
  atomicAdd(&out[(size_t)d * HID + k],
            p * (qkvs[(size_t)s * 512 + 256 + k] + eout[(size_t)e * HID + k]));
}

// ======================================================================
extern "C" void kernel_launch(void* const* d_in, const int* in_sizes, int n_in,
                              void* d_out, int out_size, void* d_ws, size_t ws_size,
                              hipStream_t stream) {
  (void)in_sizes; (void)n_in; (void)out_size;
  const int* event_type_ids = (const int*)d_in[0];
  const int* src_ids        = (const int*)d_in[1];
  const float* src_mask     = (const float*)d_in[2];
  const int* dst_ids        = (const int*)d_in[3];
  const float* dst_mask     = (const float*)d_in[4];
  const int* event_edge_ids = (const int*)d_in[5];
  const float* event_embs   = (const float*)d_in[6];
  const float* event_mask   = (const float*)d_in[7];
  const float* event_ts     = (const float*)d_in[8];
  const int* node_ids       = (const int*)d_in[9];
  const int* edge_ids       = (const int*)d_in[10];
  const int* edge_index     = (const int*)d_in[11];
  const int* time_scalar    = (const int*)d_in[12];
  const float* memory       = (const float*)d_in[13];
  const float* last_update  = (const float*)d_in[14];
  const float* node_feats   = (const float*)d_in[15];
  const float* edge_feats   = (const float*)d_in[16];
  const float* time_w       = (const float*)d_in[17];
  const float* time_b       = (const float*)d_in[18];
  const float* gru_w_ih     = (const float*)d_in[19];
  const float* gru_w_hh     = (const float*)d_in[20];
  const float* gru_b_ih     = (const float*)d_in[21];
  const float* gru_b_hh     = (const float*)d_in[22];
  const float* wq = (const float*)d_in[23]; const float* bq = (const float*)d_in[24];
  const float* wk = (const float*)d_in[25]; const float* bk = (const float*)d_in[26];
  const float* wv = (const float*)d_in[27]; const float* bv = (const float*)d_in[28];
  const float* we = (const float*)d_in[29];
  const float* wskip = (const float*)d_in[30]; const float* bskip = (const float*)d_in[31];
  float* out = (float*)d_out;

  // ---- workspace carve-out (~340 MB total) ----
  char* ws = (char*)d_ws;
  size_t off = 0;
  auto alloc = [&](size_t bytes) -> void* {
    void* p = ws + off;
    off += (bytes + 255) & ~(size_t)255;
    return p;
  };
  float* mem2 = (float*)alloc((size_t)MAXN * HID * 4);
  const size_t map_start = off;
  unsigned* map_src = (unsigned*)alloc((size_t)MAXN * 4);
  unsigned* map_dst = (unsigned*)alloc((size_t)MAXN * 4);
  const size_t map_bytes = off - map_start;
  const size_t zero_start = off;
  float* cnt_src = (float*)alloc((size_t)EVT * 4);
  float* cnt_dst = (float*)alloc((size_t)EVT * 4);
  float* agg_src = (float*)alloc((size_t)EVT * 640 * 4);
  float* agg_dst = (float*)alloc((size_t)EVT * 640 * 4);
  float* denom   = (float*)alloc((size_t)NNODE * 4);
  unsigned* amax = (unsigned*)alloc((size_t)NNODE * 4);
  const size_t zero_bytes = off - zero_start;
  float* xh    = (float*)alloc((size_t)2 * EVT * 768 * 4);
  float* wcat  = (float*)alloc((size_t)512 * 768 * 4);
  float* g     = (float*)alloc((size_t)2 * EVT * 512 * 4);
  float* newv  = (float*)alloc((size_t)2 * EVT * HID * 4);
  float* xnode = (float*)alloc((size_t)NNODE * HID * 4);
  float* wqkvs = (float*)alloc((size_t)512 * HID * 4);
  float* bqkvs = (float*)alloc((size_t)512 * 4);
  float* qkvs  = (float*)alloc((size_t)NNODE * 512 * 4);
  float* eout  = (float*)alloc((size_t)NEDGE * HID * 4);
  float* alpha = (float*)alloc((size_t)NEDGE * 4);
  float* exv   = (float*)alloc((size_t)NEDGE * 4);
  (void)ws_size;

  // ---- pipeline ----
  hipMemcpyAsync(mem2, memory, (size_t)MAXN * HID * 4, hipMemcpyDeviceToDevice, stream);
  hipMemsetAsync(ws + map_start, 0xFF, map_bytes, stream);   // maps = UINT_MAX
  hipMemsetAsync(ws + zero_start, 0, zero_bytes, stream);    // counters/accumulators

  k_assign<<<(EVT + 255) / 256, 256, 0, stream>>>(src_ids, dst_ids, map_src, map_dst);
  k_scatter_msgs<<<EVT, HID, 0, stream>>>(event_type_ids, src_ids, src_mask, dst_ids, dst_mask,
                                          event_edge_ids, event_embs, event_mask, event_ts,
                                          last_update, memory, time_w, time_b, map_src, map_dst,
                                          agg_src, agg_dst, cnt_src, cnt_dst);
  k_build_wcat<<<dim3(512, 3), 256, 0, stream>>>(gru_w_ih, gru_w_hh, wcat);
  k_build_xh<<<dim3(2 * EVT, 3), 256, 0, stream>>>(src_ids, dst_ids, map_src, map_dst,
                                                   agg_src, agg_dst, cnt_src, cnt_dst, memory, xh);
  // GRU GEMM: (8192 x 768) @ (768 x 512)^T
  k_gemm_wmma<4><<<dim3(2 * EVT / 16, 512 / 64), 32, 0, stream>>>(xh, wcat, nullptr, g,
                                                                  2 * EVT, 512, 768);
  k_gru_gate<<<2 * EVT, HID, 0, stream>>>(g, xh, gru_b_ih, gru_b_hh, newv);
  k_scatter_set<<<EVT, HID, 0, stream>>>(src_ids, newv, mem2);                      // src first
  k_scatter_set<<<EVT, HID, 0, stream>>>(dst_ids, newv + (size_t)EVT * HID, mem2);  // dst overrides
  k_node_x<<<NNODE, HID, 0, stream>>>(node_ids, node_feats, mem2, xnode);
  k_build_wqkvs<<<512, HID, 0, stream>>>(wq, bq, wk, bk, wv, bv, wskip, bskip, wqkvs, bqkvs);
  // q/k/v/skip GEMM: (20000 x 128) @ (128 x 512)^T + bias
  k_gemm_wmma<4><<<dim3(NNODE / 16, 512 / 64), 32, 0, stream>>>(xnode, wqkvs, bqkvs, qkvs,
                                                                NNODE, 512, 128);
  // edge projection GEMM: (320000 x 256) @ (256 x 128)^T, attr built in LDS
  k_edge_proj_wmma<<<NEDGE / 16, 32, 0, stream>>>(edge_ids, last_update, edge_feats, time_w,
                                                  time_b, we, time_scalar, eout);
  k_alpha<<<NEDGE / 8, 256, 0, stream>>>(edge_index, qkvs, eout, alpha, amax);
  k_exp_denom<<<(NEDGE + 255) / 256, 256, 0, stream>>>(edge_index, alpha, amax, exv, denom);
  k_skip_init<<<NNODE, HID, 0, stream>>>(qkvs, out);
  k_out<<<NEDGE, HID, 0, stream>>>(edge_index, qkvs, eout, exv, denom, out);
}